// MultiBoxLoss_47425028882766
// MI455X (gfx1250) — compile-verified
//
#include <hip/hip_runtime.h>
#include <hip/hip_bf16.h>

typedef __attribute__((ext_vector_type(2))) float v2f;
typedef __attribute__((ext_vector_type(8))) float v8f;
typedef __attribute__((ext_vector_type(4))) unsigned int v4u;
typedef __attribute__((ext_vector_type(8))) unsigned int v8u;

#define C_THRESH 0.35f
#define C_LOWGT  0.2f

__device__ __forceinline__ float sl1(float x) {
    float a = fabsf(x);
    return a < 1.f ? 0.5f * a * a : a - 0.5f;
}

// Wave-level sum of two independent quantities using one V_WMMA_F32_16X16X4_F32.
// A is a row-selector (row0 picks K=0,2 ; row1 picks K=1,3), B carries per-lane data.
// => D[0][n] = va(n)+va(16+n), D[1][n] = vb(n)+vb(16+n).
// Caller adds c[0],c[1] over lanes 0..15 to get wave totals.
__device__ __forceinline__ void wave_sum2_wmma(float va, float vb, float* o0, float* o1) {
    int lane = threadIdx.x & 31;
    v2f a;
    a.x = ((lane & 15) == 0) ? 1.f : 0.f;
    a.y = ((lane & 15) == 1) ? 1.f : 0.f;
    v2f bm;
    bm.x = va;
    bm.y = vb;
    v8f c = {};
#if __has_builtin(__builtin_amdgcn_wmma_f32_16x16x4_f32)
    c = __builtin_amdgcn_wmma_f32_16x16x4_f32(false, a, false, bm, (short)0, c, false, false);
    *o0 = c[0];
    *o1 = c[1];
#else
    float s0 = va, s1 = vb;
    for (int o = 16; o > 0; o >>= 1) {
        s0 += __shfl_down(s0, o);
        s1 += __shfl_down(s1, o);
    }
    *o0 = (lane == 0) ? s0 : 0.f;
    *o1 = (lane == 0) ? s1 : 0.f;
#endif
}

// Issue a TDM DMA of `nelem` f32 from `gsrc` into LDS at byte offset `lds_off`,
// then wait on TENSORcnt. Call from one wave only; EXEC is ignored by TDM.
__device__ __forceinline__ void tdm_load_f32_to_lds(const float* gsrc, unsigned lds_off,
                                                    unsigned nelem) {
    unsigned long long ga = (unsigned long long)gsrc;
    v4u g0;
    g0.x = 1u;                 // count=1, user descriptor, no gather
    g0.y = lds_off;            // LDS byte address
    g0.z = (unsigned)ga;       // global_addr[31:0]
    g0.w = (unsigned)((ga >> 32) & 0x01FFFFFFull) | (2u << 30);  // addr[56:32] | type=2
    v8u g1;
    g1[0] = 0x00020000u;                         // wg_mask=0, data_size=4B, no flags
    g1[1] = (nelem & 0xFFFFu) << 16;             // tensor_dim0[15:0] @ bits 63:48
    g1[2] = (nelem >> 16) | (1u << 16);          // tensor_dim0[31:16], tensor_dim1=1
    g1[3] = (nelem & 0xFFFFu) << 16;             // tensor_dim1[31:16]=0, tile_dim0=nelem
    g1[4] = 0u;                                  // tile_dim1=0, tile_dim2=0 (unused)
    g1[5] = nelem;                               // tensor_dim0_stride[31:0]
    g1[6] = 0u;                                  // stride hi, tensor_dim1_stride lo
    g1[7] = 0u;
    asm volatile("tensor_load_to_lds %0, %1" ::"s"(g0), "s"(g1) : "memory");
#if __has_builtin(__builtin_amdgcn_s_wait_tensorcnt)
    __builtin_amdgcn_s_wait_tensorcnt(0);
#else
    asm volatile("s_wait_tensorcnt 0x0" ::: "memory");
#endif
}

// K1: per-(b,p) IoU against all truths; per-prior best-truth; per-truth best-prior
// via packed u64 (iou_bits<<32)|~p so atomicMax ties break toward lowest prior idx.
// Targets slab is brought into LDS by the Tensor Data Mover (async, TENSORcnt).
__global__ void mbl_k1_match(const float* __restrict__ priors, const float* __restrict__ targets,
                             float* __restrict__ bto, int* __restrict__ bti,
                             unsigned long long* __restrict__ bpo, int P, int G) {
    __shared__ float s_tg[128 * 15];
    __shared__ unsigned long long s_best[128];
    const int b = blockIdx.y;
    const int p = blockIdx.x * blockDim.x + threadIdx.x;
    const bool ok = p < P;
    const int pp = ok ? p : 0;
    const int Gc = G < 128 ? G : 128;

    for (int g = threadIdx.x; g < Gc; g += blockDim.x) s_best[g] = 0ull;
    if (threadIdx.x < 32) {  // wave 0 issues the DMA and waits on TENSORcnt
        tdm_load_f32_to_lds(targets + (size_t)b * G * 15,
                            (unsigned)(unsigned long long)&s_tg[0], (unsigned)(Gc * 15));
    }
    __syncthreads();

    float pcx = priors[pp * 4 + 0], pcy = priors[pp * 4 + 1];
    float pw = priors[pp * 4 + 2], ph = priors[pp * 4 + 3];
    float pf0 = pcx - pw * 0.5f, pf1 = pcy - ph * 0.5f;
    float pf2 = pcx + pw * 0.5f, pf3 = pcy + ph * 0.5f;
    float areap = pw * ph;
    float bestv = -1.f;
    int bestg = 0;
    const int lane = threadIdx.x & 31;
    for (int g = 0; g < Gc; ++g) {
        float t0 = s_tg[g * 15 + 0], t1 = s_tg[g * 15 + 1];
        float t2 = s_tg[g * 15 + 2], t3 = s_tg[g * 15 + 3];
        float lx = fmaxf(t0, pf0), ly = fmaxf(t1, pf1);
        float rx = fminf(t2, pf2), ry = fminf(t3, pf3);
        float iw = fmaxf(rx - lx, 0.f), ih = fmaxf(ry - ly, 0.f);
        float inter = iw * ih;
        float areat = (t2 - t0) * (t3 - t1);
        float iou = inter / (areat + areap - inter);
        if (iou > bestv) { bestv = iou; bestg = g; }  // strict > == first argmax
        unsigned long long key =
            ok ? (((unsigned long long)__float_as_uint(iou) << 32) | (unsigned)(~(unsigned)p))
               : 0ull;
        for (int o = 16; o > 0; o >>= 1) {
            unsigned long long k2 = __shfl_down(key, o);
            if (k2 > key) key = k2;
        }
        if (lane == 0) atomicMax(&s_best[g], key);  // ds_max_u64
    }
    if (ok) {
        bto[(size_t)b * P + p] = bestv;
        bti[(size_t)b * P + p] = bestg;
    }
    __syncthreads();
    for (int g = threadIdx.x; g < Gc; g += blockDim.x)
        atomicMax(&bpo[(size_t)b * G + g], s_best[g]);  // global_atomic_max_u64
}

// K2: serial per-batch scatter fill (last-write-wins in g order), any_valid flag.
__global__ void mbl_k2_fill(const unsigned long long* __restrict__ bpo,
                            float* __restrict__ bto, int* __restrict__ bti,
                            unsigned* __restrict__ anyv, int P, int G) {
    if (threadIdx.x != 0) return;
    int b = blockIdx.x;
    unsigned av = 0;
    for (int g = 0; g < G; ++g) {
        unsigned long long k = bpo[(size_t)b * G + g];
        float ov = __uint_as_float((unsigned)(k >> 32));
        int pidx = (int)(~(unsigned)(k & 0xFFFFFFFFu));
        if (pidx < 0 || pidx >= P) continue;
        bool valid = ov >= C_LOWGT;
        av |= (unsigned)valid;
        if (valid) bto[(size_t)b * P + pidx] = 2.0f;
        bti[(size_t)b * P + pidx] = g;
    }
    anyv[b] = av;
}

// K3: encodings, smooth-L1 sums (WMMA wave-reduced), CE/mine, histogram, pos counts.
__global__ void mbl_k3_loss(const float* __restrict__ loc, const float* __restrict__ conf,
                            const float* __restrict__ landm, const float* __restrict__ priors,
                            const float* __restrict__ targets,
                            const float* __restrict__ bto, const int* __restrict__ bti,
                            const unsigned* __restrict__ anyv,
                            float* __restrict__ ce, float* __restrict__ mine,
                            unsigned* __restrict__ hist1, unsigned* __restrict__ npos,
                            unsigned* __restrict__ npos1, float* __restrict__ sums,
                            int P, int G) {
    const int b = blockIdx.y;
    const int p = blockIdx.x * blockDim.x + threadIdx.x;
    const bool ok = p < P;
    const int pp = ok ? p : 0;
    const size_t idx = (size_t)b * P + pp;

    __builtin_prefetch(conf + idx * 2, 0, 1);  // global_prefetch_b8

    float ov = bto[idx];
    int g = bti[idx];
    if (g < 0 || g >= G) g = 0;
    const float* t = targets + ((size_t)b * G + g) * 15;
    int conf_t = (int)t[14];
    if (ov < C_THRESH || anyv[b] == 0u) conf_t = 0;
    bool pos = conf_t != 0;
    bool pos1 = conf_t > 0;

    float pcx = priors[pp * 4 + 0], pcy = priors[pp * 4 + 1];
    float pw = priors[pp * 4 + 2], ph = priors[pp * 4 + 3];
    float vx = 0.1f * pw, vy = 0.1f * ph;

    // loc encode + smooth L1
    float t0 = t[0], t1 = t[1], t2 = t[2], t3 = t[3];
    float ecx = ((t0 + t2) * 0.5f - pcx) / vx;
    float ecy = ((t1 + t3) * 0.5f - pcy) / vy;
    float ew = logf((t2 - t0) / pw) * 5.0f;  // /VAR1
    float eh = logf((t3 - t1) / ph) * 5.0f;
    const float* ld = loc + idx * 4;
    float ll = sl1(ld[0] - ecx) + sl1(ld[1] - ecy) + sl1(ld[2] - ew) + sl1(ld[3] - eh);
    ll = (pos && ok) ? ll : 0.f;

    // landmark encode + smooth L1
    const float* lmd = landm + idx * 10;
    float lm = 0.f;
    for (int i = 0; i < 5; ++i) {
        float lx = (t[4 + 2 * i] - pcx) / vx;
        float ly = (t[5 + 2 * i] - pcy) / vy;
        lm += sl1(lmd[2 * i] - lx) + sl1(lmd[2 * i + 1] - ly);
    }
    lm = (pos1 && ok) ? lm : 0.f;

    // cross entropy over 2 classes
    const float* cd = conf + idx * 2;
    float c0 = cd[0], c1 = cd[1];
    float m = fmaxf(c0, c1);
    float lse = m + logf(expf(c0 - m) + expf(c1 - m));
    int ci = conf_t < 0 ? 0 : (conf_t > 1 ? 1 : conf_t);
    float cev = lse - (ci ? c1 : c0);
    float mv = pos ? 0.f : cev;
    if (ok) {
        ce[idx] = cev;
        mine[idx] = mv;
        atomicAdd(&hist1[(size_t)b * 512 + (__float_as_uint(mv) >> 22)], 1u);
    }

    unsigned long long mp = __ballot(pos && ok);
    unsigned long long mp1 = __ballot(pos1 && ok);
    if ((threadIdx.x & 31) == 0) {
        if (mp) atomicAdd(&npos[b], (unsigned)__popcll(mp));
        if (mp1) atomicAdd(npos1, (unsigned)__popcll(mp1));
    }

    // WMMA wave reduction of (ll, lm); EXEC is full here (whole block alive).
    float r0, r1;
    wave_sum2_wmma(ll, lm, &r0, &r1);
    __shared__ float sred[2];
    if (threadIdx.x < 2) sred[threadIdx.x] = 0.f;
    __syncthreads();
    if ((threadIdx.x & 31) < 16) {
        atomicAdd(&sred[0], r0);  // ds_add_f32
        atomicAdd(&sred[1], r1);
    }
    __syncthreads();
    if (threadIdx.x == 0) {
        atomicAdd(&sums[0], sred[0]);
        atomicAdd(&sums[1], sred[1]);
    }
}

// K4a: level-1 threshold scan (descending over 512 float-bit bins).
__global__ void mbl_k4a_scan(const unsigned* __restrict__ hist1, const unsigned* __restrict__ npos,
                             int* __restrict__ Tb, int* __restrict__ rb, int P) {
    if (threadIdx.x != 0) return;
    int b = blockIdx.x;
    long long num_neg = 7ll * (long long)npos[b];
    if (num_neg > P - 1) num_neg = P - 1;
    if (num_neg <= 0) { Tb[b] = 1 << 30; rb[b] = 0; return; }
    long long cum = 0;
    for (int i = 511; i >= 0; --i) {
        long long c = hist1[(size_t)b * 512 + i];
        if (cum + c >= num_neg) { Tb[b] = i; rb[b] = (int)(num_neg - cum); return; }
        cum += c;
    }
    Tb[b] = 0;
    rb[b] = (int)(num_neg - cum);
}

// K4b: level-2 histogram (next 12 mantissa bits) restricted to the boundary bin.
__global__ void mbl_k4b_hist2(const float* __restrict__ mine, const int* __restrict__ Tb,
                              unsigned* __restrict__ hist2, int P) {
    int b = blockIdx.y;
    int p = blockIdx.x * blockDim.x + threadIdx.x;
    if (p >= P) return;
    unsigned bits = __float_as_uint(mine[(size_t)b * P + p]);
    if ((int)(bits >> 22) == Tb[b])
        atomicAdd(&hist2[(size_t)b * 4096 + ((bits >> 10) & 0xFFFu)], 1u);
}

// K4c: level-2 threshold scan.
__global__ void mbl_k4c_scan2(const unsigned* __restrict__ hist2, const int* __restrict__ Tb,
                              const int* __restrict__ rb, int* __restrict__ T2b,
                              int* __restrict__ r2b) {
    if (threadIdx.x != 0) return;
    int b = blockIdx.x;
    if (Tb[b] >= (1 << 29)) { T2b[b] = 1 << 30; r2b[b] = 0; return; }
    int want = rb[b];
    long long cum = 0;
    for (int i = 4095; i >= 0; --i) {
        long long c = hist2[(size_t)b * 4096 + i];
        if (cum + c >= want) { T2b[b] = i; r2b[b] = (int)(want - cum); return; }
        cum += c;
    }
    T2b[b] = 0;
    r2b[b] = 0;
}

// K5: sel = pos | neg, sum CE with the WMMA wave reduction.
__global__ void mbl_k5_sel(const float* __restrict__ ce, const float* __restrict__ mine,
                           const int* __restrict__ Tb, const int* __restrict__ T2b,
                           const int* __restrict__ r2b, unsigned* __restrict__ ticket,
                           float* __restrict__ sums, int P) {
    const int b = blockIdx.y;
    const int p = blockIdx.x * blockDim.x + threadIdx.x;
    const bool ok = p < P;
    const size_t idx = (size_t)b * P + (ok ? p : 0);
    float val = 0.f;
    if (ok) {
        float mv = mine[idx];
        bool sel = false;
        if (mv == 0.f) {
            sel = true;  // pos (or zero-CE neg: contributes 0 either way)
        } else {
            unsigned bits = __float_as_uint(mv);
            int b1 = (int)(bits >> 22);
            if (b1 > Tb[b]) {
                sel = true;
            } else if (b1 == Tb[b]) {
                int b2 = (int)((bits >> 10) & 0xFFFu);
                if (b2 > T2b[b]) {
                    sel = true;
                } else if (b2 == T2b[b]) {
                    unsigned tk = atomicAdd(&ticket[b], 1u);
                    sel = tk < (unsigned)r2b[b];
                }
            }
        }
        if (sel) val = ce[idx];
    }
    float r0, r1;
    wave_sum2_wmma(val, 0.f, &r0, &r1);
    (void)r1;
    __shared__ float sred;
    if (threadIdx.x == 0) sred = 0.f;
    __syncthreads();
    if ((threadIdx.x & 31) < 16) atomicAdd(&sred, r0);
    __syncthreads();
    if (threadIdx.x == 0) atomicAdd(&sums[2], sred);
}

// K6: final normalization, in reference tuple order.
__global__ void mbl_k6_final(const unsigned* __restrict__ npos, const unsigned* __restrict__ npos1,
                             const float* __restrict__ sums, float* __restrict__ out, int B) {
    if (blockIdx.x != 0 || threadIdx.x != 0) return;
    unsigned tot = 0;
    for (int b = 0; b < B; ++b) tot += npos[b];
    float N = fmaxf((float)tot, 1.f);
    float N1 = fmaxf((float)(*npos1), 1.f);
    out[0] = sums[0] / N;   // loss_l / N
    out[1] = sums[2] / N;   // loss_c / N
    out[2] = sums[1] / N1;  // loss_landm / N1
}

extern "C" void kernel_launch(void* const* d_in, const int* in_sizes, int n_in,
                              void* d_out, int out_size, void* d_ws, size_t ws_size,
                              hipStream_t stream) {
    const float* loc = (const float*)d_in[0];
    const float* conf = (const float*)d_in[1];
    const float* landm = (const float*)d_in[2];
    const float* priors = (const float*)d_in[3];
    const float* targets = (const float*)d_in[4];

    const int P = in_sizes[3] / 4;
    const int B = (int)((long long)in_sizes[0] / ((long long)P * 4));
    const int G = (int)((long long)in_sizes[4] / ((long long)B * 15));
    const size_t NBP = (size_t)B * (size_t)P;

    char* w = (char*)d_ws;
    size_t off = 0;
    float* bto = (float*)(w + off); off += NBP * 4;
    int* bti = (int*)(w + off); off += NBP * 4;
    float* ce = (float*)(w + off); off += NBP * 4;
    float* mine = (float*)(w + off); off += NBP * 4;
    const size_t zoff = off;  // everything below re-zeroed each launch
    unsigned long long* bpo = (unsigned long long*)(w + off); off += (size_t)B * G * 8;
    unsigned* hist1 = (unsigned*)(w + off); off += (size_t)B * 512 * 4;
    unsigned* hist2 = (unsigned*)(w + off); off += (size_t)B * 4096 * 4;
    unsigned* npos = (unsigned*)(w + off); off += (size_t)B * 4;
    unsigned* anyv = (unsigned*)(w + off); off += (size_t)B * 4;
    unsigned* ticket = (unsigned*)(w + off); off += (size_t)B * 4;
    int* Tb = (int*)(w + off); off += (size_t)B * 4;
    int* rb = (int*)(w + off); off += (size_t)B * 4;
    int* T2b = (int*)(w + off); off += (size_t)B * 4;
    int* r2b = (int*)(w + off); off += (size_t)B * 4;
    unsigned* npos1 = (unsigned*)(w + off); off += 8;
    float* sums = (float*)(w + off); off += 16;

    hipMemsetAsync(w + zoff, 0, off - zoff, stream);

    dim3 blk(256);
    dim3 grd((unsigned)((P + 255) / 256), (unsigned)B);
    mbl_k1_match<<<grd, blk, 0, stream>>>(priors, targets, bto, bti, bpo, P, G);
    mbl_k2_fill<<<B, 32, 0, stream>>>(bpo, bto, bti, anyv, P, G);
    mbl_k3_loss<<<grd, blk, 0, stream>>>(loc, conf, landm, priors, targets, bto, bti, anyv,
                                         ce, mine, hist1, npos, npos1, sums, P, G);
    mbl_k4a_scan<<<B, 32, 0, stream>>>(hist1, npos, Tb, rb, P);
    mbl_k4b_hist2<<<grd, blk, 0, stream>>>(mine, Tb, hist2, P);
    mbl_k4c_scan2<<<B, 32, 0, stream>>>(hist2, Tb, rb, T2b, r2b);
    mbl_k5_sel<<<grd, blk, 0, stream>>>(ce, mine, Tb, T2b, r2b, ticket, sums, P);
    mbl_k6_final<<<1, 32, 0, stream>>>(npos, npos1, sums, (float*)d_out, B);
}